// NCNLayer_18253611008505
// MI455X (gfx1250) — compile-verified
//
#include <hip/hip_runtime.h>
#include <hip/hip_bf16.h>
#include <math.h>

typedef __attribute__((ext_vector_type(2))) float v2f;
typedef __attribute__((ext_vector_type(8))) float v8f;

#define DFEAT 64

// ---------------------------------------------------------------- zero scratch
__global__ __launch_bounds__(256) void zero_f32(float* __restrict__ p, int n) {
    int t = blockIdx.x * 256 + threadIdx.x;
    if (t < n) p[t] = 0.0f;
}

// ---------------------------------------------------------------- fused GEMMs
// node_out = feat @ Wn + bn ; h = feat @ Wm + bm
// One wave (32 lanes) computes one 16-row tile of both outputs using
// V_WMMA_F32_16X16X4_F32, accumulating over K=64 in 16 steps of K=4.
// A 16x4 f32 layout: lanes 0-15 -> M=lane, VGPR{0,1}=K{0,1};
//                    lanes 16-31 -> M=lane-16, VGPR{0,1}=K{2,3}.
// B 4x16 f32 layout mirrors A: lanes hold N, khalf selects K pair.
__global__ __launch_bounds__(256) void gemm_wmma(
    const float* __restrict__ feat,
    const float* __restrict__ Wn, const float* __restrict__ bn,
    const float* __restrict__ Wm, const float* __restrict__ bm,
    float* __restrict__ node_out, float* __restrict__ hout, int ntiles)
{
    __shared__ float sWn[DFEAT * DFEAT];   // 16 KB
    __shared__ float sWm[DFEAT * DFEAT];   // 16 KB
    const int tid = threadIdx.x;
    for (int i = tid; i < DFEAT * DFEAT; i += 256) { sWn[i] = Wn[i]; sWm[i] = Wm[i]; }
    __syncthreads();

    const int wave = tid >> 5;
    const int lane = tid & 31;
    const int tile = blockIdx.x * 8 + wave;
    if (tile >= ntiles) return;            // wave-uniform: EXEC stays all-ones for WMMA

    const int mlane = lane & 15;
    const int khalf = lane >> 4;
    const int row0  = tile * 16;
    const int arow  = row0 + mlane;

    v8f accN[4], accM[4];
    for (int nt = 0; nt < 4; ++nt)
        for (int j = 0; j < 8; ++j) { accN[nt][j] = 0.0f; accM[nt][j] = 0.0f; }

    for (int kb = 0; kb < DFEAT; kb += 4) {
        const int ka = kb + khalf * 2;
        v2f a;
        a.x = feat[arow * DFEAT + ka];
        a.y = feat[arow * DFEAT + ka + 1];
        #pragma unroll
        for (int nt = 0; nt < 4; ++nt) {
            const int col = nt * 16 + mlane;
            v2f b;
            b.x = sWn[ka * DFEAT + col];
            b.y = sWn[(ka + 1) * DFEAT + col];
            accN[nt] = __builtin_amdgcn_wmma_f32_16x16x4_f32(
                false, a, false, b, (short)0, accN[nt], false, false);
            v2f c;
            c.x = sWm[ka * DFEAT + col];
            c.y = sWm[(ka + 1) * DFEAT + col];
            accM[nt] = __builtin_amdgcn_wmma_f32_16x16x4_f32(
                false, a, false, c, (short)0, accM[nt], false, false);
        }
    }

    // D layout: lanes 0-15 -> M=v, N=lane ; lanes 16-31 -> M=v+8, N=lane-16
    #pragma unroll
    for (int nt = 0; nt < 4; ++nt) {
        const int col = nt * 16 + mlane;
        const float bnv = bn[col], bmv = bm[col];
        #pragma unroll
        for (int v = 0; v < 8; ++v) {
            const int mrow = row0 + v + 8 * khalf;
            node_out[mrow * DFEAT + col] = accN[nt][v] + bnv;
            hout[mrow * DFEAT + col]     = accM[nt][v] + bmv;
        }
    }
}

// ---------------------------------------------------------------- edge scatter
// 16 threads per edge; each handles a float4 feature chunk.
__global__ __launch_bounds__(256) void edge_scatter(
    const int* __restrict__ src, const int* __restrict__ dst,
    const float* __restrict__ attn, const float* __restrict__ h,
    float* __restrict__ agg, float* __restrict__ deg, int n_edges)
{
    int t = blockIdx.x * 256 + threadIdx.x;
    int e = t >> 4;
    if (e >= n_edges) return;
    int c = (t & 15) * 4;
    int s = src[e], d = dst[e];
    float a = attn[e];
    const float4 v = *(const float4*)(h + (size_t)s * DFEAT + c);
    float* ap = agg + (size_t)d * DFEAT + c;
    atomicAdd(ap + 0, a * v.x);
    atomicAdd(ap + 1, a * v.y);
    atomicAdd(ap + 2, a * v.z);
    atomicAdd(ap + 3, a * v.w);
    if ((t & 15) == 0) atomicAdd(&deg[d], 1.0f);
}

// ---------------------------------------------------------------- combine + column reduction
// Block = 64 cols x 4 row-lanes, covers 64 node rows; partial sum/sumsq per
// column reduced in LDS, then one atomicAdd per column per block.
__global__ __launch_bounds__(256) void combine_reduce(
    const float* __restrict__ node_out, const float* __restrict__ h,
    const float* __restrict__ agg, const float* __restrict__ deg,
    const float* __restrict__ alpha,
    float* __restrict__ out_pre, float* __restrict__ sums, float* __restrict__ sumsq,
    int n_nodes)
{
    __shared__ float ssum[256];
    __shared__ float ssq[256];
    const int col = threadIdx.x & 63;
    const int rl  = threadIdx.x >> 6;           // 0..3
    const float a = 1.0f / (1.0f + __expf(-alpha[0]));
    const int rbase = blockIdx.x * 64;
    float ls = 0.0f, lq = 0.0f;
    for (int r = rbase + rl; r < rbase + 64; r += 4) {
        if (r < n_nodes) {
            const size_t idx = (size_t)r * DFEAT + col;
            const float hv = h[idx];
            const float cn = (deg[r] > 0.0f) ? agg[idx] : hv;
            const float v = a * node_out[idx] + (1.0f - a) * cn;
            out_pre[idx] = v;
            ls += v;
            lq += v * v;
        }
    }
    ssum[threadIdx.x] = ls;
    ssq[threadIdx.x]  = lq;
    __syncthreads();
    if (rl == 0) {
        float s = ssum[col] + ssum[64 + col] + ssum[128 + col] + ssum[192 + col];
        float q = ssq[col]  + ssq[64 + col]  + ssq[128 + col]  + ssq[192 + col];
        atomicAdd(&sums[col], s);
        atomicAdd(&sumsq[col], q);
    }
}

// ---------------------------------------------------------------- normalize + relu (in place on d_out)
__global__ __launch_bounds__(256) void normalize_relu(
    const float* __restrict__ sums, const float* __restrict__ sumsq,
    const float* __restrict__ gamma, const float* __restrict__ beta,
    float* __restrict__ out, int n_nodes)
{
    int t = blockIdx.x * 256 + threadIdx.x;
    int total = n_nodes * DFEAT;
    if (t >= total) return;
    const int col = t & 63;
    const float n = (float)n_nodes;
    const float mean = sums[col] / n;
    float var = (sumsq[col] - n * mean * mean) / (n - 1.0f);
    var = fmaxf(var, 0.0f);
    const float denom = sqrtf(var) + 1e-8f;
    const float x = (out[t] - mean) / denom;
    const float v = gamma[col] * x + beta[col];
    out[t] = fmaxf(v, 0.0f);
}

// ---------------------------------------------------------------- launch
extern "C" void kernel_launch(void* const* d_in, const int* in_sizes, int n_in,
                              void* d_out, int out_size, void* d_ws, size_t ws_size,
                              hipStream_t stream)
{
    const float* feat  = (const float*)d_in[0];
    const int*   src   = (const int*)  d_in[1];
    const int*   dst   = (const int*)  d_in[2];
    const float* attn  = (const float*)d_in[3];
    const float* Wn    = (const float*)d_in[4];
    const float* bn    = (const float*)d_in[5];
    const float* Wm    = (const float*)d_in[6];
    const float* bm    = (const float*)d_in[7];
    const float* alpha = (const float*)d_in[8];
    const float* gamma = (const float*)d_in[9];
    const float* beta  = (const float*)d_in[10];

    const int n_nodes = in_sizes[0] / DFEAT;
    const int n_edges = in_sizes[1];

    // Workspace layout (floats): node_out | h | agg | deg | sums | sumsq
    float* ws       = (float*)d_ws;
    float* node_out = ws;
    float* h        = node_out + (size_t)n_nodes * DFEAT;
    float* agg      = h        + (size_t)n_nodes * DFEAT;
    float* deg      = agg      + (size_t)n_nodes * DFEAT;
    float* sums     = deg      + n_nodes;
    float* sumsq    = sums     + DFEAT;

    // 1) zero the accumulated regions (agg, deg, sums, sumsq are contiguous)
    const int nzero = n_nodes * DFEAT + n_nodes + 2 * DFEAT;
    zero_f32<<<(nzero + 255) / 256, 256, 0, stream>>>(agg, nzero);

    // 2) fused WMMA GEMMs
    const int ntiles = (n_nodes + 15) / 16;
    gemm_wmma<<<(ntiles + 7) / 8, 256, 0, stream>>>(feat, Wn, bn, Wm, bm,
                                                    node_out, h, ntiles);

    // 3) edge gather / scatter-add
    const long ethreads = (long)n_edges * 16;
    edge_scatter<<<(int)((ethreads + 255) / 256), 256, 0, stream>>>(
        src, dst, attn, h, agg, deg, n_edges);

    // 4) gated combine + column sum/sumsq
    combine_reduce<<<(n_nodes + 63) / 64, 256, 0, stream>>>(
        node_out, h, agg, deg, alpha, (float*)d_out, sums, sumsq, n_nodes);

    // 5) z-score normalize + affine + relu, in place on d_out
    const int total = n_nodes * DFEAT;
    normalize_relu<<<(total + 255) / 256, 256, 0, stream>>>(
        sums, sumsq, gamma, beta, (float*)d_out, n_nodes);
}